// BallQuery_85358180040913
// MI455X (gfx1250) — compile-verified
//
#include <hip/hip_runtime.h>

// ---------------------------------------------------------------------------
// PointNet++ ball query + grouping for MI455X (gfx1250, wave32).
//
// Shapes: B=8, N=8192, M=2048, K=32, C=CT=64, radius=0.2.
// Outputs: neighbor_features [8,67,2048,32] ++ temb_grouped [8,64,2048,32]
//          = 275 MB float32. Inputs total ~35 MB (fits in 192MB L2).
//
// Roofline: memory bound on output writes (275MB) + gather reads (L2-hit).
// Design:
//   K1: ball query via V_WMMA_F32_16X16X4_F32.
//       d2 = A*B + C with A[m]=[-2cx,-2cy,-2cz,1], B[n]=[px,py,pz,|p|^2],
//       C[m][n]=|c_m|^2  ->  one WMMA = 16 centers x 16 points of d2.
//       Points staged to LDS in 2048-pt chunks as float4; the per-tile
//       B-fragment ds_load_b64 is software-pipelined (load t+16 before
//       consuming t) so the LDS latency overlaps WMMA + selection instead
//       of stalling on s_wait_dscnt 0.
//       Ordered first-K selection via ballot prefix-popc (low16/high16 of
//       the ballot = the two C/D-layout rows per VGPR), early exit once all
//       16 centers have K hits.
//   K2: one wave per center (lane k = neighbor k, K==32==wave32). Gathers
//       are L2-resident; outputs written with non-temporal stores so the
//       275MB write stream does not evict the gather-resident inputs.
// ---------------------------------------------------------------------------

#define BATCH 8
#define NPTS  8192
#define MCTR  2048
#define KNBR  32
#define CFEAT 64
#define CTEMB 64
#define RAD2  0.04f
#define CHUNK 2048

typedef float v2f __attribute__((ext_vector_type(2)));
typedef float v8f __attribute__((ext_vector_type(8)));

// ---------------------------- Kernel 1: ball query -------------------------
// grid = B * (M/64) = 256 blocks, 128 threads (4 waves). Wave w handles 16
// centers m0..m0+15 of batch b. LDS: 32KB point stage + 8KB idx lists.
__global__ __launch_bounds__(128) void ball_query_wmma(
    const float* __restrict__ pc,    // [B,3,N]
    const float* __restrict__ cc,    // [B,3,M]
    int* __restrict__ idx_out)       // [B,M,K]
{
    __shared__ float4 pts[CHUNK];            // (px,py,pz,|p|^2)
    __shared__ int    idxbuf[4 * 16 * KNBR]; // per-wave first-K lists

    const int wave = threadIdx.x >> 5;
    const int lane = threadIdx.x & 31;
    const int b    = blockIdx.x >> 5;                      // 32 blocks/batch
    const int m0   = ((blockIdx.x & 31) * 4 + wave) * 16;  // 64 centers/block

    const float* pcb = pc + b * 3 * NPTS;
    const float* ccb = cc + b * 3 * MCTR;

    const bool hi  = lane >= 16;
    const int  sub = lane & 15;

    // --- A fragment: lanes 0-15 hold K=0,1 (-2cx,-2cy); 16-31 K=2,3 (-2cz,1)
    float cx = ccb[0 * MCTR + m0 + sub];
    float cy = ccb[1 * MCTR + m0 + sub];
    float cz = ccb[2 * MCTR + m0 + sub];
    float cn = cx * cx + cy * cy + cz * cz;
    v2f afrag;
    afrag.x = hi ? (-2.0f * cz) : (-2.0f * cx);
    afrag.y = hi ? 1.0f         : (-2.0f * cy);

    // --- C init: C[m][n] = |c_m|^2; VGPR r holds row m=r (lanes 0-15) and
    //     m=r+8 (lanes 16-31) -> per-lane shuffle of cn.
    v8f cinit;
    const int srcbase = hi ? 8 : 0;
#pragma unroll
    for (int r = 0; r < 8; ++r)
        cinit[r] = __shfl(cn, srcbase + r, 32);

    int cnt[16];
#pragma unroll
    for (int i = 0; i < 16; ++i) cnt[i] = 0;
    int* myidx = &idxbuf[wave * 16 * KNBR];

    // B-fragment element index within the staged float2 view: the lane's
    // slice of tile t lives at float2 element t*2 + laneoff.
    const float2* qbase  = reinterpret_cast<const float2*>(pts);
    const int     laneoff = (sub << 1) | (hi ? 1 : 0);

    bool wave_done = false;
    for (int chunk = 0; chunk < NPTS; chunk += CHUNK) {
        __syncthreads();
        // Stage this chunk of points into LDS as float4. Vectorized: each
        // thread moves 4 consecutive points per step (3x b128 global loads,
        // 4x b128 LDS stores), |p|^2 precomputed once.
        for (int p = threadIdx.x * 4; p < CHUNK; p += 128 * 4) {
            float4 x = *reinterpret_cast<const float4*>(
                &pcb[0 * NPTS + chunk + p]);
            float4 y = *reinterpret_cast<const float4*>(
                &pcb[1 * NPTS + chunk + p]);
            float4 z = *reinterpret_cast<const float4*>(
                &pcb[2 * NPTS + chunk + p]);
            pts[p + 0] = make_float4(x.x, y.x, z.x,
                                     x.x * x.x + y.x * y.x + z.x * z.x);
            pts[p + 1] = make_float4(x.y, y.y, z.y,
                                     x.y * x.y + y.y * y.y + z.y * z.y);
            pts[p + 2] = make_float4(x.z, y.z, z.z,
                                     x.z * x.z + y.z * y.z + z.z * z.z);
            pts[p + 3] = make_float4(x.w, y.w, z.w,
                                     x.w * x.w + y.w * y.w + z.w * z.w);
        }
        __syncthreads();

        if (!wave_done) {
            // Software-pipelined B-fragment: preload tile 0, then inside the
            // loop issue the load for tile t+16 before consuming tile t.
            float2 vnext = qbase[laneoff];
            for (int t = 0; t < CHUNK; t += 16) {
                v2f bfrag; bfrag.x = vnext.x; bfrag.y = vnext.y;
                const int tn = (t + 16 < CHUNK) ? (t + 16) : t;
                vnext = qbase[tn * 2 + laneoff];

                // d2[m][n] for 16 centers x 16 points in one instruction.
                v8f d = __builtin_amdgcn_wmma_f32_16x16x4_f32(
                    false, afrag, false, bfrag, (short)0, cinit, false, false);

                unsigned any = 0;
#pragma unroll
                for (int r = 0; r < 8; ++r) {
                    unsigned mask = (unsigned)__ballot(d[r] < RAD2);
                    any |= mask;
                    if (mask) {
                        unsigned mlo = mask & 0xFFFFu; // row m=r
                        unsigned mhi = mask >> 16;     // row m=r+8
                        unsigned mym = hi ? mhi : mlo;
                        if ((mym >> sub) & 1u) {
                            int row   = r + (hi ? 8 : 0);
                            int cbase = hi ? cnt[r + 8] : cnt[r];
                            int slot  = cbase +
                                __popc(mym & ((1u << sub) - 1u));
                            if (slot < KNBR)
                                myidx[row * KNBR + slot] = chunk + t + sub;
                        }
                        cnt[r]     += __popc(mlo);
                        cnt[r + 8] += __popc(mhi);
                    }
                }
                if (any) {  // counts only change when something hit
                    bool full = true;
#pragma unroll
                    for (int i = 0; i < 16; ++i) full &= (cnt[i] >= KNBR);
                    if (full) { wave_done = true; break; }
                }
            }
        }
        if (__syncthreads_and((int)wave_done)) break;  // uniform early exit
    }

    // Write indices, padding slots >= count with first valid (or 0 if none).
#pragma unroll
    for (int mm = 0; mm < 16; ++mm) {
        int cm   = cnt[mm];
        int v0   = myidx[mm * KNBR];          // first valid (garbage if cm==0)
        int fill = (cm > 0) ? v0 : 0;
        int val  = myidx[mm * KNBR + lane];
        if (lane >= cm) val = fill;
        idx_out[(b * MCTR + m0 + mm) * KNBR + lane] = val;
    }
}

// ---------------------------- Kernel 2: gather -----------------------------
// One wave per center; lane k = neighbor slot k (K==32==wave32).
// Gathers hit L2 (all inputs ~35MB resident); 128B coalesced NT stores.
__global__ __launch_bounds__(256) void group_gather(
    const float* __restrict__ pc,    // [B,3,N]
    const float* __restrict__ cc,    // [B,3,M]
    const float* __restrict__ te,    // [B,CT,N]
    const float* __restrict__ pf,    // [B,C,N]
    const int*   __restrict__ idx,   // [B,M,K]
    float* __restrict__ out)
{
    const int wave = threadIdx.x >> 5;
    const int lane = threadIdx.x & 31;
    const int cl   = blockIdx.x * 8 + wave;   // 0..16383
    const int b    = cl >> 11;
    const int m    = cl & (MCTR - 1);

    const int i = idx[cl * KNBR + lane];

    float* out1 = out;                                       // [B,67,M,K]
    float* out2 = out + (size_t)BATCH * 67 * MCTR * KNBR;    // [B,64,M,K]

    const float* pcb = pc + b * 3 * NPTS;
#pragma unroll
    for (int d = 0; d < 3; ++d) {
        float c = cc[(b * 3 + d) * MCTR + m];
        float v = pcb[d * NPTS + i] - c;
        __builtin_nontemporal_store(
            v, &out1[((b * 67 + d) * MCTR + m) * KNBR + lane]);
    }
    const float* pfb = pf + b * CFEAT * NPTS;
#pragma unroll 4
    for (int c = 0; c < CFEAT; ++c) {
        float v = pfb[c * NPTS + i];
        __builtin_nontemporal_store(
            v, &out1[((b * 67 + 3 + c) * MCTR + m) * KNBR + lane]);
    }
    const float* teb = te + b * CTEMB * NPTS;
#pragma unroll 4
    for (int c = 0; c < CTEMB; ++c) {
        float v = teb[c * NPTS + i];
        __builtin_nontemporal_store(
            v, &out2[((b * CTEMB + c) * MCTR + m) * KNBR + lane]);
    }
}

// ---------------------------------------------------------------------------
extern "C" void kernel_launch(void* const* d_in, const int* in_sizes, int n_in,
                              void* d_out, int out_size, void* d_ws,
                              size_t ws_size, hipStream_t stream) {
    (void)in_sizes; (void)n_in; (void)out_size; (void)ws_size;
    const float* pc = (const float*)d_in[0];  // points_coords  [8,3,8192]
    const float* cc = (const float*)d_in[1];  // centers_coords [8,3,2048]
    const float* te = (const float*)d_in[2];  // temb           [8,64,8192]
    const float* pf = (const float*)d_in[3];  // points_features[8,64,8192]
    int* idx = (int*)d_ws;                    // [8,2048,32] = 2MB scratch

    ball_query_wmma<<<BATCH * (MCTR / 64), 128, 0, stream>>>(pc, cc, idx);
    group_gather<<<(BATCH * MCTR) / 8, 256, 0, stream>>>(pc, cc, te, pf, idx,
                                                         (float*)d_out);
}